// MambaMIL_40046275068031
// MI455X (gfx1250) — compile-verified
//
#include <hip/hip_runtime.h>
#include <hip/hip_bf16.h>
#include <math.h>

// MambaMIL forward, gfx1250 (MI455X).
//
// Key algebraic fact: the reference returns h[:, 0, :] only, and every stage
// is causal (left-padded depthwise conv) or per-position (projections, norms,
// gating, scan with h0=0 running left->right). Hence the output depends ONLY
// on cls_token + weights -- a single-token forward pass. The scan term at t=0
// reduces to y[d] = dt[d]*xc[d]*dot(B,C) + xc[d]*D[d] (A_log is dead: it only
// multiplies h0 == 0). This is exact, not an approximation.
//
// Implementation: one 1024-thread workgroup (32 wave32s), all tensors staged
// in LDS, all matvecs executed as V_WMMA_F32_16X16X4_F32 tiles (f32 WMMA keeps
// bit-level parity with the f32 reference within FMA-reassociation tolerance).

#define D_MODEL 256
#define D_INNER 512
#define D_STATE 16
#define DT_RANK 16
#define N_LAYERS 2

typedef __attribute__((ext_vector_type(2))) float v2f;
typedef __attribute__((ext_vector_type(8))) float v8f;

__device__ __forceinline__ float silu_f(float x) {
    return x / (1.0f + expf(-x));
}

__device__ __forceinline__ float softplus_f(float x) {
    return (x > 20.0f) ? x : log1pf(expf(x));
}

// out[tileBase .. tileBase+15] = W[tileBase..+15, 0..K) . vec
// A tile (16x4 f32): lanes 0-15 -> {K=k0,k1}, lanes 16-31 -> {K=k2,k3} (ISA layout).
// B (4x16 f32) = vec broadcast across all 16 columns, so every D column holds
// the matvec; D VGPR r: lane 0 = row r, lane 16 = row r+8.
__device__ __forceinline__ void wmma_matvec16(const float* __restrict__ W,
                                              const float* __restrict__ vec,   // LDS
                                              float* __restrict__ out,         // LDS
                                              int tileBase, int K, int lane) {
    v8f acc = {};
    const int m = tileBase + (lane & 15);
    const int khalf = (lane >> 4) << 1;          // 0 for lanes 0-15, 2 for 16-31
    const float* wrow = W + (size_t)m * (size_t)K;
#pragma unroll 4
    for (int k = 0; k < K; k += 4) {
        v2f a, b;
        a.x = wrow[k + khalf];
        a.y = wrow[k + khalf + 1];
        b.x = vec[k + khalf];
        b.y = vec[k + khalf + 1];
        acc = __builtin_amdgcn_wmma_f32_16x16x4_f32(false, a, false, b,
                                                    (short)0, acc, false, false);
    }
    if (lane == 0) {
#pragma unroll
        for (int r = 0; r < 8; ++r) out[tileBase + r] = acc[r];
    } else if (lane == 16) {
#pragma unroll
        for (int r = 0; r < 8; ++r) out[tileBase + 8 + r] = acc[r];
    }
}

// Sum of v over all 1024 threads (threads not contributing pass 0).
__device__ __forceinline__ float block_sum(float v, float* s_red, int tid) {
    __syncthreads();                              // protect s_red reuse
#pragma unroll
    for (int off = 16; off > 0; off >>= 1) v += __shfl_down(v, off, 32);
    if ((tid & 31) == 0) s_red[tid >> 5] = v;
    __syncthreads();
    if (tid < 32) {
        float r = s_red[tid];
#pragma unroll
        for (int off = 16; off > 0; off >>= 1) r += __shfl_down(r, off, 32);
        if (tid == 0) s_red[0] = r;
    }
    __syncthreads();
    return s_red[0];
}

__global__ __launch_bounds__(1024)
void mamba_cls_forward(const float* __restrict__ cls_token,
                       const float* __restrict__ fc1_w,
                       const float* __restrict__ fc1_b,
                       const float* __restrict__ ln_w,
                       const float* __restrict__ ln_b,
                       const float* __restrict__ in_proj_w,
                       const float* __restrict__ conv_w,
                       const float* __restrict__ conv_b,
                       const float* __restrict__ x_proj_w,
                       const float* __restrict__ dt_proj_w,
                       const float* __restrict__ dt_proj_b,
                       const float* __restrict__ D_param,
                       const float* __restrict__ out_proj_w,
                       const float* __restrict__ norm_w,
                       const float* __restrict__ norm_b,
                       float* __restrict__ out) {
    __shared__ float s_h[D_MODEL];
    __shared__ float s_hn[D_MODEL];
    __shared__ float s_xz[2 * D_INNER];          // xc | z
    __shared__ float s_xdbl[DT_RANK + 2 * D_STATE];
    __shared__ float s_dt[D_INNER];
    __shared__ float s_y[D_INNER];
    __shared__ float s_o[D_MODEL];
    __shared__ float s_red[32];
    __shared__ float s_bc;

    const int tid  = (int)threadIdx.x;
    const int lane = tid & 31;
    const int wave = tid >> 5;

    if (tid < D_MODEL) s_h[tid] = cls_token[tid];
    __syncthreads();

    // ---- fc1 + ReLU (256x256 matvec, 16 WMMA tiles on waves 0..15) ----
    if (wave < D_MODEL / 16)
        wmma_matvec16(fc1_w, s_h, s_hn, wave * 16, D_MODEL, lane);
    __syncthreads();
    if (tid < D_MODEL) {
        float v = s_hn[tid] + fc1_b[tid];
        s_h[tid] = fmaxf(v, 0.0f);
    }
    __syncthreads();

    for (int l = 0; l < N_LAYERS; ++l) {
        // ---- LayerNorm(h) -> hn ----
        float v   = (tid < D_MODEL) ? s_h[tid] : 0.0f;
        float mu  = block_sum(v, s_red, tid) * (1.0f / D_MODEL);
        float dv  = (tid < D_MODEL) ? (v - mu) : 0.0f;
        float var = block_sum(dv * dv, s_red, tid) * (1.0f / D_MODEL);
        float rs  = rsqrtf(var + 1e-5f);
        if (tid < D_MODEL)
            s_hn[tid] = dv * rs * ln_w[l * D_MODEL + tid] + ln_b[l * D_MODEL + tid];
        __syncthreads();

        // ---- in_proj (1024x256): 64 tiles, 2 per wave ----
        const float* Wi = in_proj_w + (size_t)l * (2 * D_INNER) * D_MODEL;
        wmma_matvec16(Wi, s_hn, s_xz, wave * 16, D_MODEL, lane);
        wmma_matvec16(Wi, s_hn, s_xz, D_INNER + wave * 16, D_MODEL, lane);
        __syncthreads();

        // ---- causal depthwise conv @ t=0: only tap k=3 survives; +bias, SiLU ----
        if (tid < D_INNER) {
            float xc = conv_w[((size_t)l * D_INNER + tid) * 4 + 3] * s_xz[tid]
                     + conv_b[l * D_INNER + tid];
            s_xz[tid] = silu_f(xc);
        }
        __syncthreads();

        // ---- x_proj (48x512): 3 tiles on waves 0..2 ----
        if (wave < 3)
            wmma_matvec16(x_proj_w + (size_t)l * 48 * D_INNER, s_xz, s_xdbl,
                          wave * 16, D_INNER, lane);
        __syncthreads();

        // ---- dt_proj (512x16): 32 tiles, one per wave ----
        wmma_matvec16(dt_proj_w + (size_t)l * D_INNER * DT_RANK, s_xdbl, s_dt,
                      wave * 16, DT_RANK, lane);
        if (tid == 0) {
            float bc = 0.0f;
            for (int n = 0; n < D_STATE; ++n)
                bc += s_xdbl[DT_RANK + n] * s_xdbl[DT_RANK + D_STATE + n];
            s_bc = bc;   // dot(B, C) at t=0
        }
        __syncthreads();

        // ---- SSM @ t=0 (h0=0): y = dt*xc*(B.C) + xc*D ; y *= silu(z) ----
        if (tid < D_INNER) {
            float dt = softplus_f(s_dt[tid] + dt_proj_b[l * D_INNER + tid]);
            float xc = s_xz[tid];
            float y  = dt * xc * s_bc + xc * D_param[l * D_INNER + tid];
            s_y[tid] = y * silu_f(s_xz[D_INNER + tid]);
        }
        __syncthreads();

        // ---- out_proj (256x512): 16 tiles on waves 0..15, + residual ----
        if (wave < D_MODEL / 16)
            wmma_matvec16(out_proj_w + (size_t)l * D_MODEL * D_INNER, s_y, s_o,
                          wave * 16, D_INNER, lane);
        __syncthreads();
        if (tid < D_MODEL) s_h[tid] += s_o[tid];
        __syncthreads();
    }

    // ---- final LayerNorm -> out[0..255] ----
    float v   = (tid < D_MODEL) ? s_h[tid] : 0.0f;
    float mu  = block_sum(v, s_red, tid) * (1.0f / D_MODEL);
    float dv  = (tid < D_MODEL) ? (v - mu) : 0.0f;
    float var = block_sum(dv * dv, s_red, tid) * (1.0f / D_MODEL);
    float rs  = rsqrtf(var + 1e-5f);
    if (tid < D_MODEL)
        out[tid] = dv * rs * norm_w[tid] + norm_b[tid];
}

extern "C" void kernel_launch(void* const* d_in, const int* in_sizes, int n_in,
                              void* d_out, int out_size, void* d_ws, size_t ws_size,
                              hipStream_t stream) {
    (void)in_sizes; (void)n_in; (void)d_ws; (void)ws_size; (void)out_size;
    // d_in[0]  = x        : mathematically dead (output = position 0 only, causal model)
    // d_in[12] = A_log    : mathematically dead (dA only multiplies h0 == 0 at t=0)
    const float* cls   = (const float*)d_in[1];
    const float* fc1w  = (const float*)d_in[2];
    const float* fc1b  = (const float*)d_in[3];
    const float* lnw   = (const float*)d_in[4];
    const float* lnb   = (const float*)d_in[5];
    const float* inw   = (const float*)d_in[6];
    const float* convw = (const float*)d_in[7];
    const float* convb = (const float*)d_in[8];
    const float* xpw   = (const float*)d_in[9];
    const float* dtw   = (const float*)d_in[10];
    const float* dtb   = (const float*)d_in[11];
    const float* Dp    = (const float*)d_in[13];
    const float* outw  = (const float*)d_in[14];
    const float* nw    = (const float*)d_in[15];
    const float* nb    = (const float*)d_in[16];

    mamba_cls_forward<<<dim3(1), dim3(1024), 0, stream>>>(
        cls, fc1w, fc1b, lnw, lnb, inw, convw, convb, xpw, dtw, dtb,
        Dp, outw, nw, nb, (float*)d_out);
}